// mBP_model_420906795390
// MI455X (gfx1250) — compile-verified
//
#include <hip/hip_runtime.h>
#include <hip/hip_bf16.h>

#define N_ATOMS 192
#define SR      32     // radial centers
#define SA      8      // angular radial centers
#define NT      8      // theta centers
#define FEAT_DIM 96
#define HDIM    128
#define RCUT    5.0f
#define RCUT_A  3.5f
#define PI_F    3.14159265358979323846f

typedef __attribute__((ext_vector_type(2))) float v2f;
typedef __attribute__((ext_vector_type(8))) float v8f;

// (1 + x/64)^-64 approximation of exp(-x), exactly as the reference computes it.
__device__ __forceinline__ float app_gauss(float x) {
    float a = 1.0f / (1.0f + x * (1.0f / 64.0f));
    a = a * a;  // ^2
    a = a * a;  // ^4
    a = a * a;  // ^8
    a = a * a;  // ^16
    a = a * a;  // ^32
    a = a * a;  // ^64
    return a;
}

// ---------------------------------------------------------------------------
// Kernel 1: per-atom descriptors. One workgroup (256 thr = 8 waves) per atom.
// Writes feat[N_ATOMS][HDIM] (cols 0..31 radial, 32..95 angular, 96..127 zero).
// ---------------------------------------------------------------------------
__global__ __launch_bounds__(256) void descriptor_kernel(
    const float* __restrict__ pos, const float* __restrict__ spec,
    const float* __restrict__ Rs, const float* __restrict__ Rs_ang,
    const float* __restrict__ theta_s, const float* __restrict__ width,
    const float* __restrict__ width_ang, const float* __restrict__ zeta,
    float* __restrict__ feat)
{
    __shared__ float px[N_ATOMS], py[N_ATOMS], pz[N_ATOMS], sp[N_ATOMS];
    __shared__ float sRs[SR], sRsa[SA], sct[NT], sst[NT];
    __shared__ float gradp[8][SR];        // radial partials per j-group
    __shared__ float gang[SA * NT];       // angular accumulators
    __shared__ float nbx[N_ATOMS], nby[N_ATOMS], nbz[N_ATOMS];
    __shared__ float nbr[N_ATOMS], nbw[N_ATOMS];
    __shared__ int   nb_count;

    const int tid = threadIdx.x;
    const int i   = blockIdx.x;

    if (tid < N_ATOMS) {
        px[tid] = pos[3 * tid + 0];
        py[tid] = pos[3 * tid + 1];
        pz[tid] = pos[3 * tid + 2];
        sp[tid] = spec[tid];
    }
    if (tid < SR) sRs[tid] = Rs[tid];
    if (tid >= SR && tid < SR + SA) sRsa[tid - SR] = Rs_ang[tid - SR];
    if (tid >= 64 && tid < 64 + NT) {
        float t = theta_s[tid - 64];
        sct[tid - 64] = __cosf(t);
        sst[tid - 64] = __sinf(t);
    }
    if (tid < SA * NT) gang[tid] = 0.0f;
    if (tid == 0) nb_count = 0;
    __syncthreads();

    const float w_rad = width[0];
    const float w_ang = width_ang[0];
    const float zt    = zeta[0];
    const float xi = px[i], yi = py[i], zi = pz[i];

    // ---- radial: lane s in [0,32), j-group g in [0,8) strides over j ----
    {
        const int s = tid & (SR - 1);
        const int g = tid >> 5;
        float acc = 0.0f;
        for (int j = g; j < N_ATOMS; j += 8) {
            if (j == i) continue;
            float dx = xi - px[j], dy = yi - py[j], dz = zi - pz[j];
            float r = __builtin_sqrtf(dx * dx + dy * dy + dz * dz + 1e-12f);
            float fc = 0.0f;
            if (r <= RCUT && r > 1e-8f)
                fc = 0.5f * (1.0f + __cosf(r * (PI_F / RCUT)));
            float d = r - sRs[s];
            acc += sp[j] * app_gauss(w_rad * d * d) * fc;
        }
        gradp[g][s] = acc;
    }

    // ---- angular neighbor list: j within RCUT_A of i ----
    if (tid < N_ATOMS && tid != i) {
        const int j = tid;
        float dx = xi - px[j], dy = yi - py[j], dz = zi - pz[j];
        float r = __builtin_sqrtf(dx * dx + dy * dy + dz * dz + 1e-12f);
        if (r <= RCUT_A && r > 1e-8f) {
            float fc = 0.5f * (1.0f + __cosf(r * (PI_F / RCUT_A)));
            if (fc > 0.0f) {
                int idx = atomicAdd(&nb_count, 1);
                nbx[idx] = dx; nby[idx] = dy; nbz[idx] = dz;
                nbr[idx] = r;  nbw[idx] = fc * sp[j];
            }
        }
    }
    __syncthreads();

    // ---- angular: loop over ordered pairs (j,k), j != k, 0.5 factor folded ----
    {
        const int nb = nb_count;
        float accA[SA * NT];
#pragma unroll
        for (int q = 0; q < SA * NT; ++q) accA[q] = 0.0f;
        bool any = false;

        const int npair = nb * nb;
        for (int p = tid; p < npair; p += 256) {
            const int jj = p / nb;
            const int kk = p - jj * nb;
            if (jj == kk) continue;
            any = true;

            const float rjx = nbx[jj], rjy = nby[jj], rjz = nbz[jj], rj = nbr[jj];
            const float rkx = nbx[kk], rky = nby[kk], rkz = nbz[kk], rk = nbr[kk];

            float dot = rjx * rkx + rjy * rky + rjz * rkz;
            float ct  = dot / (rj * rk + 1e-12f);
            ct = fminf(fmaxf(ct, -1.0f + 1e-6f), 1.0f - 1e-6f);
            float st  = __builtin_sqrtf(1.0f - ct * ct);
            float rbar = 0.5f * (rj + rk);
            float w = 0.5f * nbw[jj] * nbw[kk];   // 0.5 einsum prefactor folded

            float angv[NT];
#pragma unroll
            for (int t = 0; t < NT; ++t) {
                float cdt  = ct * sct[t] + st * sst[t];
                float base = fmaxf(1.0f + cdt, 1e-7f);
                // 2^(1-z) * base^z  ==  exp2((1-z) + z*log2(base))
                angv[t] = exp2f((1.0f - zt) + zt * log2f(base));
            }
#pragma unroll
            for (int s = 0; s < SA; ++s) {
                float d  = rbar - sRsa[s];
                float wr = w * app_gauss(w_ang * d * d);
#pragma unroll
                for (int t = 0; t < NT; ++t) accA[s * NT + t] += wr * angv[t];
            }
        }
        if (any) {
#pragma unroll
            for (int q = 0; q < SA * NT; ++q) atomicAdd(&gang[q], accA[q]);
        }
    }
    __syncthreads();

    // ---- write padded feature row ----
    if (tid < SR) {
        float v = 0.0f;
#pragma unroll
        for (int g = 0; g < 8; ++g) v += gradp[g][tid];
        feat[i * HDIM + tid] = v;
    } else if (tid < SR + SA * NT) {
        feat[i * HDIM + tid] = gang[tid - SR];
    } else if (tid < HDIM) {
        feat[i * HDIM + tid] = 0.0f;
    }
}

// ---------------------------------------------------------------------------
// Kernel 2: 3-layer MLP using V_WMMA_F32_16X16X4_F32 (exact fp32 matrix path).
// Grid = 12 blocks (one 16-row M tile each), 8 waves = 8 N tiles of 16 cols.
// A layout (ISA 7.12.2, 32-bit 16x4): m = lane&15, VGPR0/1 = K pair,
// lanes 16-31 carry K+2. B assumed symmetric with lanes = N columns.
// ---------------------------------------------------------------------------
__global__ __launch_bounds__(256) void mlp_kernel(
    const float* __restrict__ feat,
    const float* __restrict__ W1, const float* __restrict__ b1,
    const float* __restrict__ W2, const float* __restrict__ b2,
    const float* __restrict__ W3, const float* __restrict__ b3,
    float* __restrict__ partial)
{
    __shared__ float h1[16 * HDIM];
    __shared__ float h2[16 * HDIM];
    __shared__ float red[256];

    const int tid  = threadIdx.x;
    const int wave = tid >> 5;        // 0..7 -> N tile
    const int lane = tid & 31;
    const int half = lane >> 4;       // K sub-pair selector
    const int l16  = lane & 15;       // row (A) / col (B,C,D)
    const int m0   = blockIdx.x * 16; // M tile base row
    const int col  = wave * 16 + l16; // global output column

    __builtin_prefetch(W2, 0, 1);     // warm far cache while layer 1 runs

    // ---- layer 1: feat[16,96] @ W1[96,128] ----
    v8f c = {0.f, 0.f, 0.f, 0.f, 0.f, 0.f, 0.f, 0.f};
    for (int k0 = 0; k0 < FEAT_DIM; k0 += 4) {
        const int ka = k0 + 2 * half;
        v2f a, b;
        a.x = feat[(m0 + l16) * HDIM + ka];
        a.y = feat[(m0 + l16) * HDIM + ka + 1];
        b.x = W1[ka * HDIM + col];
        b.y = W1[(ka + 1) * HDIM + col];
        c = __builtin_amdgcn_wmma_f32_16x16x4_f32(false, a, false, b,
                                                  (short)0, c, false, false);
    }
#pragma unroll
    for (int v = 0; v < 8; ++v) {
        const int mr = v + 8 * half;  // C/D layout: VGPR v -> rows v / v+8
        h1[mr * HDIM + col] = tanhf(c[v] + b1[col]);
    }
    __syncthreads();

    // ---- layer 2: h1[16,128] @ W2[128,128] ----
    v8f c2 = {0.f, 0.f, 0.f, 0.f, 0.f, 0.f, 0.f, 0.f};
    for (int k0 = 0; k0 < HDIM; k0 += 4) {
        const int ka = k0 + 2 * half;
        v2f a, b;
        a.x = h1[l16 * HDIM + ka];
        a.y = h1[l16 * HDIM + ka + 1];
        b.x = W2[ka * HDIM + col];
        b.y = W2[(ka + 1) * HDIM + col];
        c2 = __builtin_amdgcn_wmma_f32_16x16x4_f32(false, a, false, b,
                                                   (short)0, c2, false, false);
    }
#pragma unroll
    for (int v = 0; v < 8; ++v) {
        const int mr = v + 8 * half;
        h2[mr * HDIM + col] = tanhf(c2[v] + b2[col]);
    }
    __syncthreads();

    // ---- layer 3 + sum over the 16 rows of this tile ----
    {
        const int m  = tid >> 4;      // 0..15 local row
        const int kg = tid & 15;      // k stride group
        float acc = 0.0f;
        for (int k = kg; k < HDIM; k += 16) acc += h2[m * HDIM + k] * W3[k];
        red[tid] = acc;
    }
    __syncthreads();
    for (int s = 128; s > 0; s >>= 1) {
        if (tid < s) red[tid] += red[tid + s];
        __syncthreads();
    }
    if (tid == 0) partial[blockIdx.x] = red[0] + 16.0f * b3[0];
}

// ---------------------------------------------------------------------------
// Kernel 3: deterministic final reduction of the 12 block partials.
// ---------------------------------------------------------------------------
__global__ void finalize_kernel(const float* __restrict__ partial,
                                float* __restrict__ out)
{
    if (threadIdx.x == 0) {
        float s = 0.0f;
        for (int b = 0; b < 12; ++b) s += partial[b];
        out[0] = s;
    }
}

extern "C" void kernel_launch(void* const* d_in, const int* in_sizes, int n_in,
                              void* d_out, int out_size, void* d_ws, size_t ws_size,
                              hipStream_t stream)
{
    const float* pos      = (const float*)d_in[0];
    const float* spec     = (const float*)d_in[1];
    const float* Rs       = (const float*)d_in[2];
    const float* Rs_ang   = (const float*)d_in[3];
    const float* theta_s  = (const float*)d_in[4];
    const float* width    = (const float*)d_in[5];
    const float* widtha   = (const float*)d_in[6];
    const float* zeta     = (const float*)d_in[7];
    const float* W1       = (const float*)d_in[8];
    const float* b1       = (const float*)d_in[9];
    const float* W2       = (const float*)d_in[10];
    const float* b2       = (const float*)d_in[11];
    const float* W3       = (const float*)d_in[12];
    const float* b3       = (const float*)d_in[13];

    float* feat    = (float*)d_ws;                 // [192][128]
    float* partial = feat + N_ATOMS * HDIM;        // [12]
    float* out     = (float*)d_out;

    descriptor_kernel<<<N_ATOMS, 256, 0, stream>>>(
        pos, spec, Rs, Rs_ang, theta_s, width, widtha, zeta, feat);

    mlp_kernel<<<N_ATOMS / 16, 256, 0, stream>>>(
        feat, W1, b1, W2, b2, W3, b3, partial);

    finalize_kernel<<<1, 32, 0, stream>>>(partial, out);
}